// SmoothedCrossEntropyLoss_19705309954417
// MI455X (gfx1250) — compile-verified
//
#include <hip/hip_runtime.h>
#include <cmath>
#include <stdint.h>

#define SMOOTH_ALPHA 0.154f
#define NUM_CLASSES  50257
#define N_ROWS       4096
#define BLOCK        256
#define NWAVES       (BLOCK / 32)
#define DEPTH        4            // async pipeline depth (buffers in LDS)

typedef float f32x4 __attribute__((ext_vector_type(4)));
typedef int   v4i   __attribute__((ext_vector_type(4)));

// ---- CDNA5 async memory->LDS copy (16B per lane), tracked by ASYNCcnt ----
// Builtin signature (from hipcc diagnostic): param type is generic `int4*`.
__device__ __forceinline__ void async_copy16(const f32x4* g, f32x4* l) {
#if __has_builtin(__builtin_amdgcn_global_load_async_to_lds_b128)
    __builtin_amdgcn_global_load_async_to_lds_b128(
        (v4i*)(uintptr_t)g,       // global source
        (v4i*)(uintptr_t)l,       // LDS destination
        0, 0);                    // imm offset, cpol
#else
    asm volatile("global_load_async_to_lds_b128 %0, %1, off"
                 :: "v"((uint32_t)(uintptr_t)l),
                    "v"((uint64_t)(uintptr_t)g)
                 : "memory");
#endif
}

#if __has_builtin(__builtin_amdgcn_s_wait_asynccnt)
#define WAIT_ASYNC(n) do { __builtin_amdgcn_s_wait_asynccnt(n); \
                           asm volatile("" ::: "memory"); } while (0)
#else
#define WAIT_ASYNC(n) asm volatile("s_wait_asynccnt %0" :: "n"(n) : "memory")
#endif

// Combine two (max, sumexp) partials of an online logsumexp.
// Handles (-inf, 0) identity partials correctly (exp(-inf - finite) == 0).
__device__ __forceinline__ void lse_combine(float& m, float& s, float mo, float so) {
    float mn = fmaxf(m, mo);
    s = s * __expf(m - mn) + so * __expf(mo - mn);
    m = mn;
}

__device__ __forceinline__ void lse_vec4(float& m, float& s, f32x4 v) {
    float vmax = fmaxf(fmaxf(v.x, v.y), fmaxf(v.z, v.w));
    float mn   = fmaxf(m, vmax);
    float sc   = __expf(m - mn);
    s = fmaf(s, sc, __expf(v.x - mn) + __expf(v.y - mn)
                  + __expf(v.z - mn) + __expf(v.w - mn));
    m = mn;
}

__global__ __launch_bounds__(BLOCK) void row_lse_kernel(
    const float* __restrict__ x, const int* __restrict__ tgt,
    float* __restrict__ row_loss)
{
    const int    row      = blockIdx.x;
    const size_t rowstart = (size_t)row * NUM_CLASSES;
    const float* __restrict__ rp = x + rowstart;
    const int tid = threadIdx.x;
    const int t   = tgt[row];

    __shared__ f32x4 stage[DEPTH][BLOCK];  // 16 KB async staging (of 320 KB/WGP)
    __shared__ float s_xt;                 // logit at target class (exactly one writer)
    __shared__ float s_m[NWAVES], s_s[NWAVES];

    // Row stride (50257 floats) is only 4B-aligned; peel to a 16B boundary so
    // the streamed b128 transfers are aligned.
    const int head = (4 - (int)(rowstart & 3)) & 3;
    const int nvec = (NUM_CLASSES - head) >> 2;

    float m = -INFINITY, s = 0.0f;

    // Head scalars (0..3 elements), threads 0..head-1.
    if (tid < head) {
        float v  = rp[tid];
        float mn = fmaxf(m, v);
        s = __expf(v - mn);
        m = mn;
        if (tid == t) s_xt = v;
    }

    const f32x4* __restrict__ vp = (const f32x4*)(rp + head);

    // ---- Async-to-LDS software pipeline over uniform full tiles ----
    // Tile = BLOCK float4s (4 KB). All 256 lanes active every tile, so every
    // lane issues the same async instruction stream (uniform ASYNCcnt).
    const int T   = nvec / BLOCK;          // 49 full tiles
    const int rem = nvec - T * BLOCK;      // 19..20 ragged float4s

    if (T > 0)  async_copy16(vp + 0 * BLOCK + tid, &stage[0][tid]);
    if (T > 1)  async_copy16(vp + 1 * BLOCK + tid, &stage[1][tid]);
    if (T > 2)  async_copy16(vp + 2 * BLOCK + tid, &stage[2][tid]);

    for (int k = 0; k < T; ++k) {
        if (k + 3 < T) {
            async_copy16(vp + (size_t)(k + 3) * BLOCK + tid,
                         &stage[(k + 3) & (DEPTH - 1)][tid]);
            WAIT_ASYNC(3);                 // in-order completion -> tile k ready
        } else if (k + 3 == T) {
            WAIT_ASYNC(2);
        } else if (k + 2 == T) {
            WAIT_ASYNC(1);
        } else {
            WAIT_ASYNC(0);
        }

        f32x4 v = *(volatile f32x4*)&stage[k & (DEPTH - 1)][tid];

        const int idx = k * BLOCK + tid;               // float4 index in row
        const int rel = t - (head + (idx << 2));
        if (rel >= 0 && rel < 4) s_xt = v[rel];

        lse_vec4(m, s, v);
    }

    // Ragged float4 remainder: direct non-temporal b128 loads.
    if (tid < rem) {
        const int idx = T * BLOCK + tid;
        f32x4 v = __builtin_nontemporal_load(vp + idx);
        const int rel = t - (head + (idx << 2));
        if (rel >= 0 && rel < 4) s_xt = v[rel];
        lse_vec4(m, s, v);
    }

    // Tail scalars (0..3 elements).
    const int tail = (NUM_CLASSES - head) & 3;
    if (tid < tail) {
        const int e  = head + (nvec << 2) + tid;
        float v  = rp[e];
        float mn = fmaxf(m, v);
        s = s * __expf(m - mn) + __expf(v - mn);
        m = mn;
        if (e == t) s_xt = v;
    }

    // wave32 butterfly reduction of (m, s).
    #pragma unroll
    for (int off = 16; off >= 1; off >>= 1) {
        float mo = __shfl_xor(m, off, 32);
        float so = __shfl_xor(s, off, 32);
        lse_combine(m, s, mo, so);
    }
    if ((tid & 31) == 0) { s_m[tid >> 5] = m; s_s[tid >> 5] = s; }
    __syncthreads();

    if (tid == 0) {
        float M = s_m[0], S = s_s[0];
        #pragma unroll
        for (int w = 1; w < NWAVES; ++w) lse_combine(M, S, s_m[w], s_s[w]);
        // -log p_t = m + log(sum exp(x - m)) - x_t
        row_loss[row] = (M + __logf(S)) - s_xt;
    }
}

__global__ __launch_bounds__(BLOCK) void final_reduce_kernel(
    const float* __restrict__ row_loss, float* __restrict__ out)
{
    float acc = 0.0f;
    for (int i = threadIdx.x; i < N_ROWS; i += BLOCK)
        acc += row_loss[i];

    #pragma unroll
    for (int off = 16; off >= 1; off >>= 1)
        acc += __shfl_xor(acc, off, 32);

    __shared__ float w[NWAVES];
    if ((threadIdx.x & 31) == 0) w[threadIdx.x >> 5] = acc;
    __syncthreads();

    if (threadIdx.x == 0) {
        float tot = 0.0f;
        #pragma unroll
        for (int i = 0; i < NWAVES; ++i) tot += w[i];
        const float scale = 1.0f - SMOOTH_ALPHA + SMOOTH_ALPHA / (float)NUM_CLASSES;
        out[0] = tot / (float)N_ROWS - logf(scale);
    }
}

extern "C" void kernel_launch(void* const* d_in, const int* in_sizes, int n_in,
                              void* d_out, int out_size, void* d_ws, size_t ws_size,
                              hipStream_t stream) {
    const float* x   = (const float*)d_in[0];   // [N_ROWS, NUM_CLASSES] f32
    const int*   tgt = (const int*)d_in[1];     // [N_ROWS] i32
    float* row_loss  = (float*)d_ws;            // 4096 floats of scratch
    float* out       = (float*)d_out;           // scalar f32

    row_lse_kernel<<<N_ROWS, BLOCK, 0, stream>>>(x, tgt, row_loss);
    final_reduce_kernel<<<1, BLOCK, 0, stream>>>(row_loss, out);
}